// RoIPooling2d_86835648791318
// MI455X (gfx1250) — compile-verified
//
#include <hip/hip_runtime.h>
#include <float.h>

// Problem constants (fixed by the reference's setup_inputs)
#define NB   2
#define CC   256
#define HH   64
#define WW   64
#define RR   2048
#define PH   7
#define PW   7
#define SCL  0.0625f
#define BINS (PH * PW)          // 49
#define PER_ROI (CC * BINS)     // 12544 = 49 * 256  -> block of 256 never crosses an ROI

// ---------------------------------------------------------------------------
// Warm the 8 MB feature map into L2 with gfx1250 global_prefetch_b8.
// One thread per 128B cacheline; fire-and-forget (no LOADcnt impact).
// ---------------------------------------------------------------------------
__global__ void roi_warm_l2(const float* __restrict__ x, int nlines) {
    int i = blockIdx.x * blockDim.x + threadIdx.x;
    if (i < nlines) {
        __builtin_prefetch((const char*)x + (size_t)i * 128, /*rw=*/0, /*loc=*/3);
    }
}

// ---------------------------------------------------------------------------
// RoI max pooling. One thread per output element (r, c, ph, pw), tid-linear
// so stores are perfectly coalesced streaming writes. r is block-uniform so
// the ROI descriptor loads + ROI-wide math scalarize to SGPRs.
// ---------------------------------------------------------------------------
__global__ void roi_pool_kernel(const float* __restrict__ x,
                                const float* __restrict__ rois,
                                float* __restrict__ out) {
    const int r = blockIdx.x / BINS;                        // block-uniform ROI id
    const int v = (blockIdx.x % BINS) * 256 + threadIdx.x;  // 0 .. PER_ROI-1
    // inner ordering matches output layout: c major, then ph, then pw
    const int c  = v / BINS;
    const int pb = v % BINS;
    const int ph = pb / PW;
    const int pw = pb % PW;

    // ROI descriptor (scalarized: r uniform within the block)
    const float* roi = rois + (size_t)r * 5;
    const int b  = (int)roi[0];
    const int xs = (int)rintf(roi[1] * SCL);   // v_rndne_f32: round-half-even == jnp.round
    const int ys = (int)rintf(roi[2] * SCL);
    const int xe = (int)rintf(roi[3] * SCL);
    const int ye = (int)rintf(roi[4] * SCL);

    const float roi_w = (float)max(xe - xs + 1, 1);
    const float roi_h = (float)max(ye - ys + 1, 1);
    const float bin_w = roi_w / (float)PW;
    const float bin_h = roi_h / (float)PH;

    // bin edges, fp32 math identical to the reference, clipped to [0, W]/[0, H]
    int wstart = min(max((int)floorf((float)pw * bin_w) + xs, 0), WW);
    int wend   = min(max((int)ceilf ((float)(pw + 1) * bin_w) + xs, 0), WW);
    int hstart = min(max((int)floorf((float)ph * bin_h) + ys, 0), HH);
    int hend   = min(max((int)ceilf ((float)(ph + 1) * bin_h) + ys, 0), HH);

    const bool empty = (hend <= hstart) || (wend <= wstart);

    const float* feat = x + ((size_t)b * CC + (size_t)c) * (HH * WW);
    float m = -FLT_MAX;                         // == jnp.finfo(f32).min
    for (int h = hstart; h < hend; ++h) {
        const float* row = feat + h * WW;
        for (int w = wstart; w < wend; ++w) {
            m = fmaxf(m, row[w]);               // L2-resident gathers
        }
    }

    const float result = empty ? 0.0f : m;
    // 103 MB streaming output: NT store, keep L2 for the feature map
    __builtin_nontemporal_store(result, out + (size_t)r * PER_ROI + v);
}

// ---------------------------------------------------------------------------
extern "C" void kernel_launch(void* const* d_in, const int* in_sizes, int n_in,
                              void* d_out, int out_size, void* d_ws, size_t ws_size,
                              hipStream_t stream) {
    const float* x    = (const float*)d_in[0];   // [2,256,64,64] f32
    const float* rois = (const float*)d_in[1];   // [2048,5] f32
    float* out        = (float*)d_out;           // [2048,256,7,7] f32

    // L2 warm-up: one prefetch per 128B line of the 8 MB feature map
    const size_t xbytes = (size_t)in_sizes[0] * sizeof(float);
    const int nlines = (int)((xbytes + 127) / 128);
    roi_warm_l2<<<(nlines + 255) / 256, 256, 0, stream>>>(x, nlines);

    // One 256-thread block per (roi, 256-output slice); 2048*49 = 100352 blocks
    roi_pool_kernel<<<RR * BINS, 256, 0, stream>>>(x, rois, out);
}